// Correlation_87969520156733
// MI455X (gfx1250) — compile-verified
//
#include <hip/hip_runtime.h>

typedef __attribute__((ext_vector_type(2))) float f32x2;
typedef __attribute__((ext_vector_type(8))) float f32x8;

#define C_    512
#define H_    96
#define W_    96
#define HW_   (H_ * W_)            // 9216
#define CHW_  ((size_t)C_ * HW_)   // 4718592
#define TILES 7                    // ceil(96/14)
#define TSTRIDE 14                 // output pixels per 16x16 tile

// One wave (32 lanes) computes a 14-pixel output tile for one (b, h):
//   A  = x1[b, :, h, p .. p+15]            (16 x 512)
//   Bw = x2[b, :, h+dy-1, p-1 .. p+14]     (16 x 512), dy = 0,1,2
//   G_dy = A . Bw^T  via chained V_WMMA_F32_16X16X4_F32 (K chunks of 4)
//   out[b, dx*3+dy, h, p+m] = G_dy[m, m+dx] / 512   for m = 0..13, dx = 0..2
__global__ __launch_bounds__(256) void corr_wmma_f32(const float* __restrict__ x1,
                                                     const float* __restrict__ x2,
                                                     float* __restrict__ out)
{
    const int lane = threadIdx.x & 31;
    const int wave = threadIdx.x >> 5;
    const int gw   = blockIdx.x * 8 + wave;        // 0 .. 5375

    const int tile = gw % TILES;
    const int h    = (gw / TILES) % H_;
    const int b    = gw / (TILES * H_);

    const int half = lane >> 4;                    // 0: K=0,1 / M,N 0..  1: K=2,3
    const int li   = lane & 15;                    // M (for A) / N (for B/C/D)
    const int p    = tile * TSTRIDE;               // first output pixel of tile

    // ---- A operand: pixel p+li (clamped; garbage rows never stored) ----
    int wA = p + li; if (wA > W_ - 1) wA = W_ - 1;

    // ---- B operand: column p-1+li (clamped; OOB columns zeroed at store) ----
    int colB = p - 1 + li;
    int colBc = colB < 0 ? 0 : (colB > W_ - 1 ? W_ - 1 : colB);

    // x2 rows h-1, h, h+1 (clamped; OOB rows zeroed at store)
    const int  h0 = h - 1, h2 = h + 1;
    const bool row0Ok = (h0 >= 0);
    const bool row2Ok = (h2 <= H_ - 1);
    const int  h0c = row0Ok ? h0 : 0;
    const int  h2c = row2Ok ? h2 : 0;

    const size_t chanOfs = (size_t)(2 * half) * HW_;   // this lane's first channel
    const float* pA  = x1 + (size_t)b * CHW_ + chanOfs + (size_t)h  * W_ + wA;
    const float* pB0 = x2 + (size_t)b * CHW_ + chanOfs + (size_t)h0c * W_ + colBc;
    const float* pB1 = x2 + (size_t)b * CHW_ + chanOfs + (size_t)h  * W_ + colBc;
    const float* pB2 = x2 + (size_t)b * CHW_ + chanOfs + (size_t)h2c * W_ + colBc;

    f32x8 acc0 = {};   // dy = 0
    f32x8 acc1 = {};   // dy = 1
    f32x8 acc2 = {};   // dy = 2

#pragma unroll 4
    for (int k = 0; k < C_ / 4; ++k) {
        f32x2 a, b0, b1, b2;
        a.x  = pA[0];   a.y  = pA[HW_];
        b0.x = pB0[0];  b0.y = pB0[HW_];
        b1.x = pB1[0];  b1.y = pB1[HW_];
        b2.x = pB2[0];  b2.y = pB2[HW_];
        pA  += 4 * HW_;  pB0 += 4 * HW_;  pB1 += 4 * HW_;  pB2 += 4 * HW_;

        acc0 = __builtin_amdgcn_wmma_f32_16x16x4_f32(false, a, false, b0,
                                                     (short)0, acc0, false, false);
        acc1 = __builtin_amdgcn_wmma_f32_16x16x4_f32(false, a, false, b1,
                                                     (short)0, acc1, false, false);
        acc2 = __builtin_amdgcn_wmma_f32_16x16x4_f32(false, a, false, b2,
                                                     (short)0, acc2, false, false);
    }

    // ---- scatter the 3 diagonals of each G to out, fixing up OOB with 0 ----
    const float scale = 1.0f / (float)C_;
    f32x8 accs[3] = { acc0, acc1, acc2 };
    const bool rowOk[3] = { row0Ok, true, row2Ok };

    for (int r = 0; r < 8; ++r) {
        const int m  = 8 * half + r;        // G row this lane holds in VGPR r
        const int dx = li - m;              // column n = li = m + dx
        if (dx < 0 || dx > 2) continue;
        if (m > 13) continue;               // tile owns m = 0..13 only
        const int w = p + m;
        if (w > W_ - 1) continue;           // last tile trim
        const int wcol  = w + dx - 1;
        const bool cOk  = (wcol >= 0) && (wcol <= W_ - 1);
        for (int dy = 0; dy < 3; ++dy) {
            const float g   = accs[dy][r];
            const float val = (cOk && rowOk[dy]) ? g * scale : 0.0f;
            out[((size_t)b * 9 + (dx * 3 + dy)) * HW_ + (size_t)h * W_ + w] = val;
        }
    }
}

extern "C" void kernel_launch(void* const* d_in, const int* in_sizes, int n_in,
                              void* d_out, int out_size, void* d_ws, size_t ws_size,
                              hipStream_t stream) {
    const float* x1 = (const float*)d_in[0];
    const float* x2 = (const float*)d_in[1];
    float* out = (float*)d_out;
    (void)in_sizes; (void)n_in; (void)out_size; (void)d_ws; (void)ws_size;

    // 8 batches * 96 rows * 7 tiles = 5376 waves; 8 waves per 256-thread block
    const int blocks = (8 * H_ * TILES) / 8;   // 672
    corr_wmma_f32<<<blocks, 256, 0, stream>>>(x1, x2, out);
}